// BoltzmannRouter_84619445666056
// MI455X (gfx1250) — compile-verified
//
#include <hip/hip_runtime.h>

// ---------------------------------------------------------------------------
// Boltzmann MoE router for MI455X (gfx1250, wave32).
//   scores = (x @ gate_w^T) / e ; probs = softmax(scores)
//   keep top-44 of 64 experts; renormalize with +1e-8 eps.
// GEMM uses V_WMMA_F32_16X16X4_F32 (full fp32 fidelity). Memory-bound on
// streaming x (134MB @ 23.3TB/s ~ 5.8us floor); W staged in LDS per K-chunk.
// ---------------------------------------------------------------------------

typedef __attribute__((ext_vector_type(2))) float v2f;
typedef __attribute__((ext_vector_type(8))) float v8f;

#define DIM        2048
#define NEXP       64
#define KC         128           // K-chunk staged in LDS
#define WSTRIDE    132           // padded LDS row stride (floats); 132%64==4 -> conflict-free
#define SSTRIDE    65            // padded per-token score stride (floats)
#define TOK_PER_WG 128           // 8 waves x 16 tokens
#define BLOCK      256
#define N_REMOVE   20            // 64 - 44
#define INV_T      0.36787944117144233f  // 1/e

__launch_bounds__(BLOCK)
__global__ void boltzmann_router_kernel(const float* __restrict__ x,
                                        const float* __restrict__ gate_w,
                                        float* __restrict__ out,
                                        int tokens) {
    // W-chunk buffer (64 x WSTRIDE floats) aliased later as score buffer
    // (TOK_PER_WG x SSTRIDE = 8320 floats <= 8448).
    __shared__ float lds[NEXP * WSTRIDE];

    const int tid  = threadIdx.x;
    const int wave = tid >> 5;           // 0..7
    const int lane = tid & 31;
    const int lm   = lane & 15;
    const int hi   = lane >> 4;          // 0 for lanes 0-15, 1 for 16-31

    const int tok0 = blockIdx.x * TOK_PER_WG;
    if (tok0 >= tokens) return;

    // A-matrix per-lane base: 16x4 f32 A layout -> lane holds
    // x[row][k + 2*hi], x[row][k + 1 + 2*hi] per k-step.
    const int row = tok0 + wave * 16 + lm;
    const float* __restrict__ aptr = x + (size_t)row * DIM + 2 * hi;

    // B-fragment LDS offsets for the four 16-expert N-tiles (mirrored layout).
    int boff[4];
#pragma unroll
    for (int n = 0; n < 4; ++n) boff[n] = (n * 16 + lm) * WSTRIDE + 2 * hi;

    const v8f vzero = {0.f, 0.f, 0.f, 0.f, 0.f, 0.f, 0.f, 0.f};
    v8f acc[4] = {vzero, vzero, vzero, vzero};

    for (int kc = 0; kc < DIM; kc += KC) {
        __syncthreads();   // all waves done reading previous chunk
        // Cooperative load of W[0:64, kc:kc+KC] -> LDS (float4, coalesced).
#pragma unroll
        for (int i = 0; i < (NEXP * KC) / (BLOCK * 4); ++i) {  // 8 iters
            int c  = (tid + i * BLOCK) * 4;      // flat float index in chunk
            int e  = c >> 7;                     // / KC
            int kk = c & (KC - 1);
            float4 v = *(const float4*)(gate_w + (size_t)e * DIM + kc + kk);
            *(float4*)(&lds[e * WSTRIDE + kk]) = v;
        }
        __syncthreads();

#pragma unroll 4
        for (int kk = 0; kk < KC; kk += 4) {
            v2f a  = *(const v2f*)(aptr + kc + kk);          // global b64
            v2f b0 = *(const v2f*)(&lds[boff[0] + kk]);      // LDS b64 (no conflicts)
            v2f b1 = *(const v2f*)(&lds[boff[1] + kk]);
            v2f b2 = *(const v2f*)(&lds[boff[2] + kk]);
            v2f b3 = *(const v2f*)(&lds[boff[3] + kk]);
            acc[0] = __builtin_amdgcn_wmma_f32_16x16x4_f32(false, a, false, b0,
                                                           (short)0, acc[0], false, false);
            acc[1] = __builtin_amdgcn_wmma_f32_16x16x4_f32(false, a, false, b1,
                                                           (short)0, acc[1], false, false);
            acc[2] = __builtin_amdgcn_wmma_f32_16x16x4_f32(false, a, false, b2,
                                                           (short)0, acc[2], false, false);
            acc[3] = __builtin_amdgcn_wmma_f32_16x16x4_f32(false, a, false, b3,
                                                           (short)0, acc[3], false, false);
        }
    }
    __syncthreads();   // everyone done with W buffer; reuse LDS for scores

    // Scatter scores: C layout -> VGPR j holds M = 8*hi + j (N = lm + 16*n).
#pragma unroll
    for (int n = 0; n < 4; ++n)
#pragma unroll
        for (int j = 0; j < 8; ++j) {
            int tl = wave * 16 + hi * 8 + j;                  // local token
            lds[tl * SSTRIDE + n * 16 + lm] = acc[n][j] * INV_T;
        }
    __syncthreads();

    // -------- Phase 2: per-token softmax + top-44 + renormalize ----------
    if (tid < TOK_PER_WG) {
        float* __restrict__ s = &lds[tid * SSTRIDE];

        float m = -3.4e38f;
        for (int i = 0; i < NEXP; ++i) m = fmaxf(m, s[i]);

        float z = 0.f;
        for (int i = 0; i < NEXP; ++i) {
            float p = expf(s[i] - m);   // un-normalized prob
            s[i] = p;
            z += p;
        }

        // Remove the 20 smallest (ties: remove later index first, matching
        // jax top_k's earliest-index preference for the kept set).
        unsigned long long rem = 0ull;
        for (int it = 0; it < N_REMOVE; ++it) {
            float vmin = 3.4e38f;
            int idx = 0;
            for (int i = 0; i < NEXP; ++i) {
                if (!((rem >> i) & 1ull)) {
                    float v = s[i];
                    if (v <= vmin) { vmin = v; idx = i; }
                }
            }
            rem |= (1ull << idx);
        }

        float ksum = 0.f;
        for (int i = 0; i < NEXP; ++i)
            if (!((rem >> i) & 1ull)) ksum += s[i];

        // weights = (p/z) / (ksum/z + eps) == p / (ksum + z*eps)
        float dinv = 1.0f / (ksum + z * 1e-8f);

        float* __restrict__ o = out + (size_t)(tok0 + tid) * NEXP;
#pragma unroll
        for (int i = 0; i < NEXP; i += 4) {
            float4 v;
            v.x = ((rem >> (i + 0)) & 1ull) ? 0.f : s[i + 0] * dinv;
            v.y = ((rem >> (i + 1)) & 1ull) ? 0.f : s[i + 1] * dinv;
            v.z = ((rem >> (i + 2)) & 1ull) ? 0.f : s[i + 2] * dinv;
            v.w = ((rem >> (i + 3)) & 1ull) ? 0.f : s[i + 3] * dinv;
            *(float4*)(o + i) = v;
        }
    }
}

extern "C" void kernel_launch(void* const* d_in, const int* in_sizes, int n_in,
                              void* d_out, int out_size, void* d_ws, size_t ws_size,
                              hipStream_t stream) {
    (void)n_in; (void)out_size; (void)d_ws; (void)ws_size;
    const float* x  = (const float*)d_in[0];   // (B,S,D) fp32
    const float* gw = (const float*)d_in[1];   // (E,D)   fp32
    float* out = (float*)d_out;                // (B,S,E) fp32

    const int tokens = in_sizes[0] / DIM;      // 16384
    const int grid   = tokens / TOK_PER_WG;    // 128
    hipLaunchKernelGGL(boltzmann_router_kernel, dim3(grid), dim3(BLOCK), 0, stream,
                       x, gw, out, tokens);
}